// SuperlatticeHamiltonian_50792283243137
// MI455X (gfx1250) — compile-verified
//
#include <hip/hip_runtime.h>

typedef float v2f __attribute__((ext_vector_type(2)));
typedef float v4f __attribute__((ext_vector_type(4)));
typedef float v8f __attribute__((ext_vector_type(8)));

namespace {
constexpr int kB   = 16;     // batch
constexpr int kS   = 4096;   // seq len
constexpr int kD   = 2048;   // embed dim
constexpr int kNS  = 6;      // number of scales
constexpr int kCH  = 32;     // S-chunks for phase-1 reduction
constexpr int kRows = kS / kCH;  // 128 rows per chunk
constexpr int kRes = 256;
constexpr int kD4  = kD / 4; // 512 float4 per row
constexpr int kU   = 16;     // K-steps batched per outer iteration in K3

struct Ptrs6 { const float* p[6]; };
} // namespace

__device__ __forceinline__ int n_of_scale(int sc) {
  // n = min(SCALES[i], 8) for SCALES = {2,3,5,8,13,21}
  return sc == 0 ? 2 : (sc == 1 ? 3 : (sc == 2 ? 5 : 8));
}

__device__ __forceinline__ int pos_to_idx(int s) {
  // positions = linspace(-1,1,4096); idx = clip(int((pos+1)/2*255), 0, 255)
  float pos = -1.0f + 2.0f * (float)s / 4095.0f;
  float t = (pos + 1.0f) * 0.5f * 255.0f;
  int i = (int)t;
  return i < 0 ? 0 : (i > 255 ? 255 : i);
}

// ---------------------------------------------------------------------------
// K1: streaming partial sums over S. 512 blocks x 256 threads.
// Each block owns (batch b, chunk ch of 128 rows); each thread accumulates
// two float4 columns with non-temporal b128 loads (512 MiB use-once stream,
// NT hint keeps it from sweeping the 192 MB L2). This kernel is the roofline:
// ~512 MiB / 23.3 TB/s ~= 23 us; everything downstream touches < 5 MiB.
// ---------------------------------------------------------------------------
__global__ void partial_sum_kernel(const float* __restrict__ x,
                                   float* __restrict__ part) {
  const int b  = blockIdx.x >> 5;   // /kCH
  const int ch = blockIdx.x & 31;
  const int t  = threadIdx.x;       // 0..255
  const v4f* xv = (const v4f*)x;
  v4f acc0 = {0.f, 0.f, 0.f, 0.f};
  v4f acc1 = {0.f, 0.f, 0.f, 0.f};
  long base = ((long)b * kS + (long)ch * kRows) * kD4;
  for (int s = 0; s < kRows; ++s) {
    v4f a = __builtin_nontemporal_load(xv + base + t);
    v4f c = __builtin_nontemporal_load(xv + base + t + 256);
    acc0 += a;
    acc1 += c;
    base += kD4;
  }
  v4f* pv = (v4f*)part;
  long pbase = ((long)b * kCH + ch) * kD4;
  pv[pbase + t]       = acc0;
  pv[pbase + t + 256] = acc1;
}

// ---------------------------------------------------------------------------
// K2: reduce 32 chunk partials per (b,d) column -> pooled = mean over S.
// ---------------------------------------------------------------------------
__global__ void pooled_kernel(const float* __restrict__ part,
                              float* __restrict__ pooled) {
  const int g = blockIdx.x * 256 + threadIdx.x;  // 0..32767
  const int b = g >> 11;
  const int d = g & 2047;
  float s = 0.f;
  for (int ch = 0; ch < kCH; ++ch)
    s += part[((long)(b * kCH + ch)) * kD + d];
  pooled[g] = s * (1.0f / (float)kS);
}

// ---------------------------------------------------------------------------
// K3: logits = pooled @ W^T via V_WMMA_F32_16X16X4_F32 (M=16 batches,
// N=16 (units zero-padded), K=2048), then bias + masked 16-lane softmax.
// One full wave per scale; 6 waves; EXEC stays all-ones for the whole loop.
// Latency strategy: only ~1-2 waves/SIMD here, so instead of a 512-long
// serial {load -> wait -> wmma} chain, each outer iteration issues 32
// independent b64 loads (16 K-steps of A and B into register arrays), waits
// once, then runs 16 back-to-back WMMAs: 32 stalls instead of 512.
// Zero-padding of N to 16 is branch-free: clamped (in-bounds) row pointer
// + 0/1 mask multiply, so EXEC never changes around the WMMAs.
// A frag (16x4 f32, ISA 7.12.2): lane<16 holds M=lane, K={k0,k0+1};
//                                lane>=16 holds M=lane-16, K={k0+2,k0+3}.
// B frag (4x16): lane<16 holds N=lane, K rows {k0,k0+1}; upper half K+2.
// D frag: vgpr j, lane<16 -> (M=j, N=lane); lane>=16 -> (M=j+8, N=lane-16).
// ---------------------------------------------------------------------------
__global__ void amp_kernel(const float* __restrict__ pooled, Ptrs6 Ws, Ptrs6 bs,
                           float* __restrict__ amp_ws) {
  const int wave = threadIdx.x >> 5;      // scale index 0..5
  const int lane = threadIdx.x & 31;
  const int col  = lane & 15;             // A: M (batch) / B: N (unit)
  const int koff = (lane >> 4) << 1;      // upper half-wave holds K+2
  const int n = n_of_scale(wave);
  const float* W    = Ws.p[wave];
  const float* bias = bs.p[wave];

  const int   rowc = col < n ? col : (n - 1);   // clamped: always in-bounds
  const float msk  = col < n ? 1.0f : 0.0f;     // zero pad units n..15

  v8f c = {0.f, 0.f, 0.f, 0.f, 0.f, 0.f, 0.f, 0.f};
  const float* arow = pooled + col * kD + koff;   // pooled[batch=col][k]
  const float* brow = W + rowc * kD + koff;       // W[unit=rowc][k]
  for (int k0 = 0; k0 < kD; k0 += 4 * kU) {
    v2f ab[kU], bb[kU];
#pragma unroll
    for (int u = 0; u < kU; ++u) {               // 32 independent loads
      ab[u] = *(const v2f*)(arow + k0 + 4 * u);
      bb[u] = *(const v2f*)(brow + k0 + 4 * u);
    }
#pragma unroll
    for (int u = 0; u < kU; ++u) {               // 16 back-to-back WMMAs
      v2f bm = bb[u];
      bm.x *= msk;
      bm.y *= msk;
      c = __builtin_amdgcn_wmma_f32_16x16x4_f32(false, ab[u], false, bm,
                                                (short)0, c, false, false);
    }
  }

  const float bv = bias[rowc];              // in-bounds; masked below anyway
  const int mbase = (lane >> 4) << 3;       // 0 or 8
  for (int j = 0; j < 8; ++j) {
    // logits for batch m = mbase + j live in c[j], unit = col across 16 lanes
    float v = (col < n) ? (c[j] + bv) : -3.0e38f;
    float m = v;
    m = fmaxf(m, __shfl_xor(m, 1, 32));
    m = fmaxf(m, __shfl_xor(m, 2, 32));
    m = fmaxf(m, __shfl_xor(m, 4, 32));
    m = fmaxf(m, __shfl_xor(m, 8, 32));
    float e = expf(v - m) * msk;            // invalid units -> exactly 0
    float sum = e;
    sum += __shfl_xor(sum, 1, 32);
    sum += __shfl_xor(sum, 2, 32);
    sum += __shfl_xor(sum, 4, 32);
    sum += __shfl_xor(sum, 8, 32);
    amp_ws[wave * 256 + (mbase + j) * 16 + col] = e / sum;
  }
}

// ---------------------------------------------------------------------------
// K4: out[b, scale, s] = amp @ basis[:, idx]. One 16x16 output tile per wave
// (M=16 batches, N=16 seq positions, K=8 zero-padded units) -> 2 WMMA/tile.
// 1536 tiles = 192 blocks x 8 waves. Same branch-free clamp+mask padding.
// ---------------------------------------------------------------------------
__global__ void synth_kernel(const float* __restrict__ amp_ws, Ptrs6 bases,
                             float* __restrict__ out) {
  const int wave  = threadIdx.x >> 5;
  const int lane  = threadIdx.x & 31;
  const int tile  = blockIdx.x * 8 + wave;   // 0..1535
  const int scale = tile >> 8;               // /256
  const int s0    = (tile & 255) << 4;
  const int col   = lane & 15;
  const int koff  = (lane >> 4) << 1;
  const int n = n_of_scale(scale);
  const float* amp   = amp_ws + scale * 256; // amp[batch][unit], 16x16
  const float* basis = bases.p[scale];       // basis[unit][256]

  const int s  = s0 + col;
  const int id = pos_to_idx(s);

  v8f c = {0.f, 0.f, 0.f, 0.f, 0.f, 0.f, 0.f, 0.f};
#pragma unroll
  for (int k0 = 0; k0 < 8; k0 += 4) {
    const int kr  = k0 + koff;
    const int kc0 = kr     < n ? kr     : (n - 1);  // clamped rows, in-bounds
    const int kc1 = kr + 1 < n ? kr + 1 : (n - 1);
    const float m0 = kr     < n ? 1.0f : 0.0f;
    const float m1 = kr + 1 < n ? 1.0f : 0.0f;
    v2f a = *(const v2f*)(amp + col * 16 + kr);  // A[M=col][K=kr..kr+1]
    v2f bm;                                       // B[K][N=col] = basis[k][idx]
    bm.x = basis[kc0 * kRes + id] * m0;
    bm.y = basis[kc1 * kRes + id] * m1;
    c = __builtin_amdgcn_wmma_f32_16x16x4_f32(false, a, false, bm,
                                              (short)0, c, false, false);
  }

  const int mbase = (lane >> 4) << 3;        // batch row offset 0 or 8
#pragma unroll
  for (int j = 0; j < 8; ++j) {
    // out flat index: (b * 6 + scale) * 4096 + s
    out[((mbase + j) * kNS + scale) * kS + s] = c[j];
  }
}

extern "C" void kernel_launch(void* const* d_in, const int* in_sizes, int n_in,
                              void* d_out, int out_size, void* d_ws, size_t ws_size,
                              hipStream_t stream) {
  // Input order: x, W0..W5, b0..b5, basis0..basis5 (all fp32)
  const float* x = (const float*)d_in[0];
  Ptrs6 Ws, Bs, Bas;
  for (int i = 0; i < 6; ++i) {
    Ws.p[i]  = (const float*)d_in[1 + i];
    Bs.p[i]  = (const float*)d_in[7 + i];
    Bas.p[i] = (const float*)d_in[13 + i];
  }
  float* ws     = (float*)d_ws;
  float* part   = ws;                                   // 16*32*2048 floats (4 MiB)
  float* pooled = ws + (long)kB * kCH * kD;             // +32768 floats
  float* amp    = pooled + (long)kB * kD;               // +1536 floats
  float* out    = (float*)d_out;                        // [16, 6, 4096] fp32

  partial_sum_kernel<<<kB * kCH, 256, 0, stream>>>(x, part);          // 512 blocks
  pooled_kernel<<<(kB * kD) / 256, 256, 0, stream>>>(part, pooled);   // 128 blocks
  amp_kernel<<<1, 192, 0, stream>>>(pooled, Ws, Bs, amp);             // 6 waves
  synth_kernel<<<(kNS * 256) / 8, 256, 0, stream>>>(amp, Bas, out);   // 192 blocks
}